// OverlapLoss_63110249447560
// MI455X (gfx1250) — compile-verified
//
#include <hip/hip_runtime.h>
#include <hip/hip_bf16.h>

// OverlapLoss for B=16384 layouts of S=256 tokens (128 boxes at odd slots).
// Kernel 1: one block per batch; async-copy the 128 raw boxes to LDS, build
// inverse permutation in LDS, compute IoU(child, parent)/area(child), reduce
// to per-batch partial (iou_sum, legal_cnt) in d_ws.
// Kernel 2: single block; deterministic fold of 16384 partials, final 256-value
// sum done with a chained V_WMMA_F32_16X16X4_F32 (A = ones) accumulator.

typedef __attribute__((ext_vector_type(2))) float v2f;
typedef __attribute__((ext_vector_type(8))) float v8f;

// Types for the async-to-LDS builtin: param 1 is int4* in global AS(1),
// param 2 is the LDS destination in AS(3) (per hipcc diagnostic).
typedef __attribute__((__vector_size__(4 * sizeof(int)))) int v4i_t;
typedef __attribute__((address_space(1))) v4i_t* gv4i_p;
typedef __attribute__((address_space(3))) v4i_t* lv4i_p;

#define IMG_W 1440.0f
#define IMG_H 2560.0f

#if __has_builtin(__builtin_amdgcn_global_load_async_to_lds_b128)
#define HAVE_ASYNC_LDS 1
#else
#define HAVE_ASYNC_LDS 0
#endif

__device__ __forceinline__ void wait_async_zero() {
#if __has_builtin(__builtin_amdgcn_s_wait_asynccnt)
  __builtin_amdgcn_s_wait_asynccnt(0);
#else
  asm volatile("s_wait_asynccnt 0" ::: "memory");
#endif
}

__device__ __forceinline__ void cxcywh_to_xyxy(const float4 r, float& x1, float& y1,
                                               float& x2, float& y2) {
  const float cx = r.x * IMG_W;
  const float cy = r.y * IMG_H;
  const float hw = r.z * (IMG_W * 0.5f);
  const float hh = r.w * (IMG_H * 0.5f);
  x1 = cx - hw; y1 = cy - hh; x2 = cx + hw; y2 = cy + hh;
}

__global__ __launch_bounds__(128) void overlap_batch_kernel(
    const float4* __restrict__ pred_boxes,  // [B*S] float4 (cx,cy,w,h)
    const int* __restrict__ id_arr,         // [B*S]
    const int* __restrict__ parent_arr,     // [B*S]
    float* __restrict__ part_iou,           // [B]
    float* __restrict__ part_cnt)           // [B]
{
  __shared__ float4 sbox[128];   // raw cxcywh for the 128 odd-slot boxes
  __shared__ int    sinv[128];   // inverse permutation: sinv[id] = local index
  __shared__ int    spid[128];   // parent id (local) by local index
  __shared__ float  wiou[4];
  __shared__ float  wcnt[4];

  const int b = blockIdx.x;
  const int i = threadIdx.x;                 // local index in [0,128)
  const size_t g = (size_t)b * 256 + 2 * i + 1;  // odd token slot

  // Stage the raw box tile directly into LDS (async DMA path on gfx1250).
#if HAVE_ASYNC_LDS
  __builtin_amdgcn_global_load_async_to_lds_b128(
      (gv4i_p)(v4i_t*)(void*)(pred_boxes + g),
      (lv4i_p)(v4i_t*)(void*)&sbox[i],
      /*offset=*/0, /*cpol=*/0);
#else
  sbox[i] = pred_boxes[g];
#endif

  const int myid  = id_arr[g];      // permutation value in [0,128)
  const int mypid = parent_arr[g];  // parent local id in [0,128)
  spid[i]    = mypid;
  sinv[myid] = i;

#if HAVE_ASYNC_LDS
  wait_async_zero();                // this wave's async copies landed in LDS
#endif
  __syncthreads();                  // all waves' LDS writes visible

  // Thread u handles unique id u of this layout.
  const int u = i;
  const int t = sinv[u];            // index of this id's (only) occurrence
  const int p = spid[t];            // its parent id
  const int tp = sinv[p];           // index of the parent box

  float px1, py1, px2, py2, tx1, ty1, tx2, ty2;
  cxcywh_to_xyxy(sbox[t],  px1, py1, px2, py2);
  cxcywh_to_xyxy(sbox[tp], tx1, ty1, tx2, ty2);

  const float xl = fmaxf(px1, tx1);
  const float yt = fmaxf(py1, ty1);
  const float xr = fminf(px2, tx2);
  const float yb = fminf(py2, ty2);
  const bool legal = (xr >= xl) & (yb >= yt) & (u != 0);

  const float inter = (xr - xl) * (yb - yt);
  const float area1 = (px2 - px1) * (py2 - py1);
  float iou = legal ? (inter / area1) : 0.0f;
  float cnt = legal ? 1.0f : 0.0f;

  // Deterministic wave32 xor-tree reduce, then fixed-order cross-wave fold.
  for (int off = 16; off > 0; off >>= 1) {
    iou += __shfl_xor(iou, off, 32);
    cnt += __shfl_xor(cnt, off, 32);
  }
  if ((i & 31) == 0) { wiou[i >> 5] = iou; wcnt[i >> 5] = cnt; }
  __syncthreads();
  if (i == 0) {
    part_iou[b] = ((wiou[0] + wiou[1]) + wiou[2]) + wiou[3];
    part_cnt[b] = ((wcnt[0] + wcnt[1]) + wcnt[2]) + wcnt[3];
  }
}

__global__ __launch_bounds__(256) void overlap_reduce_kernel(
    const float* __restrict__ part_iou,
    const float* __restrict__ part_cnt,
    float* __restrict__ out, int nb, float n_total)
{
  __shared__ float siou[256];
  __shared__ float scnt[256];
  const int t = threadIdx.x;

  float a = 0.0f, c = 0.0f;
  for (int k = t; k < nb; k += 256) {  // fixed order -> deterministic
    a += part_iou[k];
    c += part_cnt[k];
  }
  siou[t] = a;
  scnt[t] = c;
  __syncthreads();

  if (t < 32) {  // wave 0 only; EXEC is all-1s inside this wave
    // Sum the 256 LDS partials with a chained WMMA: D += ones(16x4) x B(4x16).
    // Each iteration folds a 64-value chunk into 16 column sums.
    v2f ones; ones[0] = 1.0f; ones[1] = 1.0f;
    v8f acc = {};
    const int col     = t & 15;
    const int rowbase = (t >> 4) << 1;  // lanes 0-15 -> rows {0,1}; 16-31 -> {2,3}
    for (int it = 0; it < 4; ++it) {
      const float* chunk = &siou[it * 64];
      v2f bv;
      bv[0] = chunk[rowbase * 16 + col];        // B[rowbase  ][col]
      bv[1] = chunk[(rowbase + 1) * 16 + col];  // B[rowbase+1][col]
      acc = __builtin_amdgcn_wmma_f32_16x16x4_f32(
          /*neg_a=*/false, ones, /*neg_b=*/false, bv,
          /*c_mod=*/(short)0, acc, /*reuse_a=*/false, /*reuse_b=*/false);
    }
    // D[M=0][N=j] lives in VGPR0 (acc[0]) on lane j (j = 0..15).
    const float r = acc[0];
    float tot = 0.0f;
    for (int j = 0; j < 16; ++j) tot += __shfl(r, j, 32);

    if (t == 0) {
      float cnt = 0.0f;
      for (int k = 0; k < 256; ++k) cnt += scnt[k];
      const float mean = tot / fmaxf(cnt, 1.0f);
      const float total_iou =
          ((cnt == 0.0f) || (mean < 0.0f) || (mean > 1.0f)) ? 0.0f : tot;
      out[0] = n_total - total_iou;
    }
  }
}

extern "C" void kernel_launch(void* const* d_in, const int* in_sizes, int n_in,
                              void* d_out, int out_size, void* d_ws, size_t ws_size,
                              hipStream_t stream) {
  const float4* pred = (const float4*)d_in[0];  // [B,S,4] f32
  const int* idv     = (const int*)d_in[1];     // [B,S] i32
  const int* pidv    = (const int*)d_in[2];     // [B,S] i32
  // d_in[3] = type_id : unused by the reference math (never read -> saves BW)
  (void)n_in; (void)out_size; (void)ws_size;

  const int nb = in_sizes[1] / 256;             // number of layouts (16384)
  float* part_iou = (float*)d_ws;               // [nb]
  float* part_cnt = part_iou + nb;              // [nb]

  overlap_batch_kernel<<<nb, 128, 0, stream>>>(pred, idv, pidv, part_iou, part_cnt);

  const float n_total = (float)((long long)nb * 128);  // N = B*HALF
  overlap_reduce_kernel<<<1, 256, 0, stream>>>(part_iou, part_cnt,
                                               (float*)d_out, nb, n_total);
}